// SparseLinear_18519898980707
// MI455X (gfx1250) — compile-verified
//
#include <hip/hip_runtime.h>
#include <hip/hip_bf16.h>

typedef __attribute__((ext_vector_type(2))) float v2f;
typedef __attribute__((ext_vector_type(8))) float v8f;

// ---------------------------------------------------------------------------
// WMMA-based 16x16 tile transpose with fused bias:
//   dst[c][r] = src[r][c] + (bias ? bias[r] : 0)     src: R x C, dst: C x R
// One wave32 per full 16x16 source tile.
// Trick: load tile T into the B-operand layout (lane = column n, VGPRs = K
// rows), multiply by identity in the A-operand layout (lane = row m), and the
// result D = I*T = T lands in the C/D layout (lane = N column, VGPR = M row).
// Each lane then owns 8 *consecutive-in-M* elements of column n, i.e. 8
// consecutive elements of row n of T^T -> two coalesced float4 stores.
// Bias rides along as the C accumulator (constant across lanes/N).
// ---------------------------------------------------------------------------
__global__ __launch_bounds__(256) void wmma_transpose_bias_kernel(
    const float* __restrict__ src, float* __restrict__ dst,
    const float* __restrict__ bias, int R, int C, int tilesR, int tilesC)
{
    int wave = (int)(blockIdx.x * (blockDim.x >> 5)) + (int)(threadIdx.x >> 5);
    if (wave >= tilesR * tilesC) return;          // wave-uniform: EXEC stays all-1s
    int lane = (int)(threadIdx.x & 31);
    int nlo  = lane & 15;                          // M for A-operand, N for B/D
    int half = lane >> 4;

    int tr = wave / tilesC;
    int tc = wave - tr * tilesC;
    int r0 = tr << 4;
    int c0 = tc << 4;

    // Accumulator C[m][n] = bias[r0+m] (VGPR v holds m = v + 8*half)
    v8f acc = {};
    if (bias) {
        const float4* bp = reinterpret_cast<const float4*>(bias + r0 + (half ? 8 : 0));
        float4 b0 = bp[0], b1 = bp[1];
        acc[0] = b0.x; acc[1] = b0.y; acc[2] = b0.z; acc[3] = b0.w;
        acc[4] = b1.x; acc[5] = b1.y; acc[6] = b1.z; acc[7] = b1.w;
    }

    // D = sum_j  I[:,4j..4j+3] (16x4, A-layout)  x  T[4j..4j+3,:] (4x16, B-layout)
#pragma unroll
    for (int j = 0; j < 4; ++j) {
        int ka = 4 * j + (half ? 2 : 0);           // K index in VGPR0 for this lane
        v2f a, b;
        // A: identity columns; lane's M is nlo.  VGPR0 -> K=ka, VGPR1 -> K=ka+1
        a.x = (nlo == ka    ) ? 1.0f : 0.0f;
        a.y = (nlo == ka + 1) ? 1.0f : 0.0f;
        // B: rows ka, ka+1 of the tile; lane's N is nlo
        const float* s0 = src + (size_t)(r0 + ka) * C + (c0 + nlo);
        b.x = s0[0];
        b.y = s0[C];
        acc = __builtin_amdgcn_wmma_f32_16x16x4_f32(
            false, a, false, b, (short)0, acc, false, false);
    }

    // Store: lane n writes dst row (c0+n), source rows r0+8*half .. +7
    float* dp = dst + (size_t)(c0 + nlo) * R + r0 + (half ? 8 : 0);
    reinterpret_cast<float4*>(dp)[0] = make_float4(acc[0], acc[1], acc[2], acc[3]);
    reinterpret_cast<float4*>(dp)[1] = make_float4(acc[4], acc[5], acc[6], acc[7]);
}

// Generic scalar transpose for edge strips (only launched if R or C % 16 != 0;
// never needed for B=512, IN/OUT=50000).
__global__ void transpose_edge_kernel(const float* __restrict__ src,
                                      float* __restrict__ dst,
                                      const float* __restrict__ bias,
                                      int R, int C, int fullR, int fullC)
{
    size_t total = (size_t)R * C;
    for (size_t i = (size_t)blockIdx.x * blockDim.x + threadIdx.x; i < total;
         i += (size_t)gridDim.x * blockDim.x) {
        int r = (int)(i / C), c = (int)(i - (size_t)r * C);
        if (r >= fullR || c >= fullC) {
            float v = src[i] + (bias ? bias[r] : 0.0f);
            dst[(size_t)c * R + r] = v;
        }
    }
}

// ---------------------------------------------------------------------------
// Zero-fill (float4 vectorized)
// ---------------------------------------------------------------------------
__global__ __launch_bounds__(256) void fill_zero_kernel(float* __restrict__ p, size_t n)
{
    size_t n4 = n >> 2;
    size_t i  = (size_t)blockIdx.x * blockDim.x + threadIdx.x;
    if (i < n4) reinterpret_cast<float4*>(p)[i] = make_float4(0.f, 0.f, 0.f, 0.f);
    if (i == 0) for (size_t t = n4 << 2; t < n; ++t) p[t] = 0.f;
}

__device__ __forceinline__ void atomAddF32(float* p, float v)
{
#if defined(__HIP_DEVICE_COMPILE__)
    unsafeAtomicAdd(p, v);     // native global_atomic_add_f32, no return
#else
    atomicAdd(p, v);
#endif
}

// ---------------------------------------------------------------------------
// COO scatter: lane = batch index.  x_t is (IN, B) so each nnz touches one
// contiguous 2KB row (coalesced loads, L2-resident); accumulation into the
// contiguous y_t row via no-return f32 atomics (resolved in L2).
// ---------------------------------------------------------------------------
#define NNZ_PER_BLOCK 8

__global__ __launch_bounds__(512) void spmm_scatter_kernel(
    const int* __restrict__ rows, const int* __restrict__ cols,
    const float* __restrict__ vals, const float* __restrict__ xt,
    float* __restrict__ yt, int nnz, int Bn)
{
    int e0 = (int)blockIdx.x * NNZ_PER_BLOCK;

    int   rr[NNZ_PER_BLOCK];
    int   cc[NNZ_PER_BLOCK];
    float vv[NNZ_PER_BLOCK];
#pragma unroll
    for (int i = 0; i < NNZ_PER_BLOCK; ++i) {
        int e = e0 + i;
        rr[i] = (e < nnz) ? rows[e] : 0;           // uniform -> scalar loads
        cc[i] = (e < nnz) ? cols[e] : 0;
        vv[i] = (e < nnz) ? vals[e] : 0.0f;
    }
    // Warm the gather rows into cache (global_prefetch_b8)
#pragma unroll
    for (int i = 0; i < NNZ_PER_BLOCK; ++i)
        __builtin_prefetch(xt + (size_t)cc[i] * Bn + threadIdx.x, 0, 0);

#pragma unroll
    for (int i = 0; i < NNZ_PER_BLOCK; ++i) {
        int e = e0 + i;
        if (e >= nnz) break;
        const float* xr = xt + (size_t)cc[i] * Bn;
        float*       yr = yt + (size_t)rr[i] * Bn;
        float        v  = vv[i];
        for (int b = (int)threadIdx.x; b < Bn; b += (int)blockDim.x)
            atomAddF32(yr + b, v * xr[b]);
    }
}

// ---------------------------------------------------------------------------
// Fallback path (workspace too small for the two transposed buffers):
// operate directly on x (B,IN) / out (B,OUT). Uncoalesced but correct.
// ---------------------------------------------------------------------------
__global__ __launch_bounds__(256) void init_out_bias_kernel(
    float* __restrict__ out, const float* __restrict__ bias, int Bn, int On)
{
    size_t total = (size_t)Bn * On;
    for (size_t i = (size_t)blockIdx.x * blockDim.x + threadIdx.x; i < total;
         i += (size_t)gridDim.x * blockDim.x)
        out[i] = bias[i % On];
}

__global__ __launch_bounds__(512) void spmm_scatter_direct_kernel(
    const int* __restrict__ rows, const int* __restrict__ cols,
    const float* __restrict__ vals, const float* __restrict__ x,
    float* __restrict__ out, int nnz, int Bn, int In, int On)
{
    int e0 = (int)blockIdx.x * NNZ_PER_BLOCK;
#pragma unroll
    for (int i = 0; i < NNZ_PER_BLOCK; ++i) {
        int e = e0 + i;
        if (e >= nnz) break;
        int r = rows[e], c = cols[e];
        float v = vals[e];
        for (int b = (int)threadIdx.x; b < Bn; b += (int)blockDim.x)
            atomAddF32(out + (size_t)b * On + r, v * x[(size_t)b * In + c]);
    }
}

// ---------------------------------------------------------------------------
extern "C" void kernel_launch(void* const* d_in, const int* in_sizes, int n_in,
                              void* d_out, int out_size, void* d_ws, size_t ws_size,
                              hipStream_t stream)
{
    const float* x    = (const float*)d_in[0];
    const int*   idx  = (const int*)d_in[1];
    const float* vals = (const float*)d_in[2];
    const float* bias = (const float*)d_in[3];

    const int nnz = in_sizes[2];
    const int On  = in_sizes[3];            // OUT (bias length)
    const int Bn  = out_size / On;          // batch
    const int In  = in_sizes[0] / Bn;       // IN

    const int* rows = idx;
    const int* cols = idx + nnz;
    float* out = (float*)d_out;

    const size_t xtBytes = (size_t)In * Bn * sizeof(float);
    const size_t ytBytes = (size_t)On * Bn * sizeof(float);

    int bdim = (Bn >= 512) ? 512 : (((Bn + 31) / 32) * 32);
    if (bdim < 32) bdim = 32;
    int scatterBlocks = (nnz + NNZ_PER_BLOCK - 1) / NNZ_PER_BLOCK;

    if (ws_size >= xtBytes + ytBytes) {
        float* xt = (float*)d_ws;
        float* yt = (float*)((char*)d_ws + xtBytes);

        // 1) x (Bn x In) -> xt (In x Bn)   [WMMA register transpose]
        {
            int tR = Bn >> 4, tC = In >> 4;
            int waves = tR * tC;
            int blocks = (waves * 32 + 255) / 256;
            wmma_transpose_bias_kernel<<<blocks, 256, 0, stream>>>(
                x, xt, nullptr, Bn, In, tR, tC);
            if ((Bn & 15) || (In & 15))
                transpose_edge_kernel<<<4096, 256, 0, stream>>>(
                    x, xt, nullptr, Bn, In, tR << 4, tC << 4);
        }
        // 2) yt = 0
        {
            size_t n = (size_t)On * Bn;
            int blocks = (int)(((n >> 2) + 255) / 256) + 1;
            fill_zero_kernel<<<blocks, 256, 0, stream>>>(yt, n);
        }
        // 3) scatter nnz contributions into yt (On x Bn)
        spmm_scatter_kernel<<<scatterBlocks, bdim, 0, stream>>>(
            rows, cols, vals, xt, yt, nnz, Bn);
        // 4) yt (On x Bn) -> out (Bn x On), + bias   [WMMA register transpose]
        {
            int tR = On >> 4, tC = Bn >> 4;
            int waves = tR * tC;
            int blocks = (waves * 32 + 255) / 256;
            wmma_transpose_bias_kernel<<<blocks, 256, 0, stream>>>(
                yt, out, bias, On, Bn, tR, tC);
            if ((On & 15) || (Bn & 15))
                transpose_edge_kernel<<<4096, 256, 0, stream>>>(
                    yt, out, bias, On, Bn, tR << 4, tC << 4);
        }
    } else {
        // Fallback: direct scatter into d_out (deterministic choice by ws_size)
        init_out_bias_kernel<<<4096, 256, 0, stream>>>(out, bias, Bn, On);
        spmm_scatter_direct_kernel<<<scatterBlocks, bdim, 0, stream>>>(
            rows, cols, vals, x, out, nnz, Bn, In, On);
    }
}